// MedSegNetV2_46411416601265
// MI455X (gfx1250) — compile-verified
//
#include <hip/hip_runtime.h>

// MedSegNetV2 texture-feature stencil for MI455X (gfx1250).
// Bandwidth-bound: ~129 MB read + 411 MB write -> ~23 us floor @ 23.3 TB/s.
// CDNA5 paths: TENSOR_LOAD_TO_LDS (TDM DMA, TENSORcnt) + NT b128 stores.

#define BB   8
#define CC   64
#define HH   224
#define WW   224
#define TH   8            // output rows per strip
#define NSTRIP (HH / TH)  // 28 strips per plane
#define LROWS (TH + 2)    // 10 halo rows staged in LDS
#define NTHREADS 448      // 14 wave32; 448 * 4 px = 224*8 px per strip

// theta = 1  =>  exp(theta*log f - 0.5*theta^2) = f * exp(-0.5)
#define EXP_NEG_HALF 0.60653065971263342f

typedef unsigned int u32x4 __attribute__((ext_vector_type(4)));
typedef int          i32x8 __attribute__((ext_vector_type(8)));
typedef int          i32x4 __attribute__((ext_vector_type(4)));
typedef float        f32x4 __attribute__((ext_vector_type(4)));

__device__ __forceinline__ float mart(float f) {
    f = fmaxf(f, 1e-5f);             // log-domain clip
    f = f * EXP_NEG_HALF;            // collapsed log/exp martingale (theta=1)
    f = fminf(fmaxf(f, 1e-4f), 1e4f);
    return f;
}

__global__ __launch_bounds__(NTHREADS) void texfeat_kernel(const float* __restrict__ x,
                                                           float* __restrict__ out) {
    __shared__ float lds[LROWS * WW];   // 10 x 224 floats = 8960 B, packed stride 224

    const int tid   = threadIdx.x;
    const int bid   = blockIdx.x;
    const int strip = bid % NSTRIP;
    const int plane = bid / NSTRIP;     // b*64 + c
    const int ty0   = strip * TH;

    const int skipY = (ty0 == 0) ? 1 : 0;       // top halo row is zero padding
    const int sy    = ty0 - 1 + skipY;          // first global row staged

    // top strip: pre-zero LDS row 0 (TDM fills rows 1..9 there)
    if (skipY && tid < WW) lds[tid] = 0.0f;

    // ---- Tensor Data Mover: one DMA stages the (10-skipY) x 224 halo strip ----
    if (tid < 32) {   // single wave issues (tensor ops ignore EXEC -> must branch)
        const unsigned long long ga =
            (unsigned long long)(size_t)(x + (size_t)plane * (HH * WW) + (size_t)sy * WW);
        const unsigned lds_base =
            (unsigned)(size_t)(&lds[0]) + (unsigned)(skipY * WW * 4);
        const unsigned tensor_d1 = (unsigned)(HH - sy);   // rows left; bottom OOB -> 0
        const unsigned tile_d1   = (unsigned)(LROWS - skipY);

        u32x4 g0;
        g0.x = 1u;                                            // count=1 (valid D#)
        g0.y = lds_base;                                      // lds_addr [63:32]
        g0.z = (unsigned)(ga & 0xFFFFFFFFull);                // global_addr lo
        g0.w = (unsigned)((ga >> 32) & 0x1FFFFFFull)          // global_addr [56:32]
               | (2u << 30);                                  // type=2 ("image")

        i32x8 g1;
        g1[0] = (int)(2u << 16);                              // data_size=4B; wg_mask=0
        g1[1] = (int)((WW & 0xFFFFu) << 16);                  // tensor_dim0 lo16
        g1[2] = (int)(((unsigned)WW >> 16) |                  // tensor_dim0 hi16
                      ((tensor_d1 & 0xFFFFu) << 16));         // tensor_dim1 lo16
        g1[3] = (int)((tensor_d1 >> 16) |                     // tensor_dim1 hi16
                      ((unsigned)WW << 16));                  // tile_dim0 = 224
        g1[4] = (int)(tile_d1 & 0xFFFFu);                     // tile_dim1; tile_dim2=0
        g1[5] = (int)WW;                                      // tensor_dim0_stride lo32
        g1[6] = 0;                                            // stride0 hi; stride1 lo
        g1[7] = 0;                                            // stride1 hi

        i32x4 gz4 = {0, 0, 0, 0};                             // 2D tile: groups 2/3 unused
        i32x8 gz8 = {0, 0, 0, 0, 0, 0, 0, 0};
        __builtin_amdgcn_tensor_load_to_lds(g0, g1, gz4, gz4, gz8, 0);
    }
    __builtin_amdgcn_s_wait_tensorcnt(0);   // issuing wave: DMA done (no-op for others)
    __syncthreads();

    // ---- compute: each thread owns 4 consecutive x-pixels of one output row ----
    const int row = tid / 56;          // 0..7
    const int seg = tid % 56;          // 0..55
    const int x0  = seg * 4;
    const int y   = ty0 + row;

    // window gx = x0-1 .. x0+4 for rows y-1..y+1 (LDS rows row..row+2)
    float w9[3][6];
    #pragma unroll
    for (int r = 0; r < 3; ++r) {
        const float* rp = &lds[(row + r) * WW];
        f32x4 a = *(const f32x4*)(rp + x0);                   // 16B aligned ds_load_b128
        float wl = rp[(x0 > 0) ? (x0 - 1) : 0];
        float wr = rp[(x0 < WW - 4) ? (x0 + 4) : (WW - 1)];
        if (x0 == 0)      wl = 0.0f;   // left zero padding
        if (x0 == WW - 4) wr = 0.0f;   // right zero padding (gx = 224)
        w9[r][0] = wl;  w9[r][1] = a.x; w9[r][2] = a.y;
        w9[r][3] = a.z; w9[r][4] = a.w; w9[r][5] = wr;
    }

    // per-element t*log(t), computed once, reused by all 4 pixels
    float el[3][6];
    #pragma unroll
    for (int r = 0; r < 3; ++r) {
        #pragma unroll
        for (int k = 0; k < 6; ++k) {
            float tv = fmaxf(w9[r][k], 1e-6f);
            el[r][k] = tv * __logf(tv);
        }
    }

    float res[4][4];   // [feature][pixel]
    #pragma unroll
    for (int j = 0; j < 4; ++j) {
        float s1 = 0.f, s2 = 0.f, se = 0.f;
        #pragma unroll
        for (int r = 0; r < 3; ++r) {
            #pragma unroll
            for (int k = 0; k < 3; ++k) {
                float v = w9[r][j + k];
                s1 += v;
                s2 = fmaf(v, v, s2);
                se += el[r][j + k];
            }
        }
        const float mean = s1 * (1.0f / 9.0f);
        const float ex2  = s2 * (1.0f / 9.0f);
        float var  = fmaxf((s2 - 9.0f * mean * mean) * (1.0f / 8.0f), 0.0f);
        float stdv = fmaxf(__fsqrt_rn(var), 1e-3f);           // torch unbiased std
        float contrast = (ex2 - mean * mean) / (stdv * stdv);
        float entropy  = -se * (1.0f / 9.0f);
        float sad = 0.f;
        #pragma unroll
        for (int r = 0; r < 3; ++r) {
            #pragma unroll
            for (int k = 0; k < 3; ++k)
                sad += fabsf(w9[r][j + k] - mean);
        }
        float homog = 1.0f / (1.0f + sad * (1.0f / 9.0f) + 1e-6f);

        res[0][j] = mart(contrast);
        res[1][j] = mart(ex2);       // energy
        res[2][j] = mart(entropy);
        res[3][j] = mart(homog);
    }

    // ---- 4 non-temporal 128-bit stores: output never re-read, don't thrash L2 ----
    #pragma unroll
    for (int f = 0; f < 4; ++f) {
        size_t off = (((size_t)plane * 4 + f) * HH + y) * WW + x0;
        f32x4 o = { res[f][0], res[f][1], res[f][2], res[f][3] };
        __builtin_nontemporal_store(o, (f32x4*)(out + off));
    }
}

extern "C" void kernel_launch(void* const* d_in, const int* in_sizes, int n_in,
                              void* d_out, int out_size, void* d_ws, size_t ws_size,
                              hipStream_t stream) {
    (void)in_sizes; (void)n_in; (void)out_size; (void)d_ws; (void)ws_size;
    const float* x = (const float*)d_in[0];
    float* out = (float*)d_out;
    const int nblocks = BB * CC * NSTRIP;   // 512 planes * 28 strips = 14336
    texfeat_kernel<<<nblocks, NTHREADS, 0, stream>>>(x, out);
}